// similarLoss1_76854144795000
// MI455X (gfx1250) — compile-verified
//
#include <hip/hip_runtime.h>

typedef float v2f __attribute__((ext_vector_type(2)));
typedef float v4f __attribute__((ext_vector_type(4)));
typedef float v8f __attribute__((ext_vector_type(8)));

#define DHW     160
#define TILE    16
#define HALO    2
#define TW      20          // TILE + 2*HALO
#define TROWS   32          // padded row count (rows 20..31 stay zero)
#define ZCHUNK  32
#define NQ      5           // I, J, I2, J2, IJ
#define INV_WIN (1.0f / 125.0f)
#define WIN_SZ  125.0f
#define N_VOXEL 8192000.0f  // 2 * 160^3

// One wave per block. Each block: 16x16 (y,x) tile, ZCHUNK z-slices, one batch.
__global__ __launch_bounds__(32) void ncc_tile_kernel(
    const float* __restrict__ real,
    const float* __restrict__ fake,
    float* __restrict__ partial)
{
    const int lane = threadIdx.x;        // 0..31
    const int l16  = lane & 15;
    const int lh   = lane >> 4;          // lane half

    const int tx0 = blockIdx.x * TILE;
    const int ty0 = blockIdx.y * TILE;
    const int nzc = DHW / ZCHUNK;        // 5
    const int batch = blockIdx.z / nzc;
    const int z0    = (blockIdx.z % nzc) * ZCHUNK;

    const size_t vbase = (size_t)batch * DHW * DHW * DHW;
    const size_t slice_stride = (size_t)DHW * DHW;

    // LDS working set (~48 KB, well under 320 KB/WGP)
    __shared__ float T[NQ][TROWS][TW];     // quantity tiles, rows 20..31 = zero pad
    __shared__ float Ut[NQ][TILE][32];     // x-pass result, TRANSPOSED [col][row], padded
    __shared__ v4f   Ring[NQ][5][64];      // z-ring of xy-sums, per-lane v4f pairs

    // one-time zero of T's pad rows (rows 20..31) and the ring
    for (int it = lane; it < NQ * (TROWS - TW) * TW; it += 32) {
        const int q   = it / ((TROWS - TW) * TW);
        const int rem = it % ((TROWS - TW) * TW);
        T[q][TW + rem / TW][rem % TW] = 0.0f;
    }
    #pragma unroll
    for (int q = 0; q < NQ; ++q)
        #pragma unroll
        for (int s = 0; s < 5; ++s) {
            Ring[q][s][lane * 2 + 0] = (v4f){0.f, 0.f, 0.f, 0.f};
            Ring[q][s][lane * 2 + 1] = (v4f){0.f, 0.f, 0.f, 0.f};
        }

    float acc[NQ][8];                     // running z-window sums (registers)
    #pragma unroll
    for (int q = 0; q < NQ; ++q)
        #pragma unroll
        for (int g = 0; g < 8; ++g) acc[q][g] = 0.0f;

    float ccsum = 0.0f;

    // steps: input slice zi = z0 - 2 + step; outputs valid for step >= 4
    for (int step = 0; step < ZCHUNK + 4; ++step) {
        const int zi = z0 - HALO + step;
        const bool zvalid = (zi >= 0) && (zi < DHW);

        __syncthreads();

        // ---- load 20x20 halo tile of this slice, build 5 quantity tiles ----
        for (int it = lane; it < TW * TW; it += 32) {
            const int tyy = it / TW;
            const int txx = it - tyy * TW;
            const int gy = ty0 - HALO + tyy;
            const int gx = tx0 - HALO + txx;
            float r = 0.0f, f = 0.0f;
            const bool inb = zvalid && (gy >= 0) && (gy < DHW) && (gx >= 0) && (gx < DHW);
            if (inb) {
                const size_t idx = vbase + ((size_t)zi * DHW + gy) * DHW + gx;
                r = real[idx];
                f = fake[idx];
                if (zi + 1 < DHW) {   // prefetch next z-slice (global_prefetch_b8)
                    __builtin_prefetch(&real[idx + slice_stride], 0, 0);
                    __builtin_prefetch(&fake[idx + slice_stride], 0, 0);
                }
            }
            T[0][tyy][txx] = r;
            T[1][tyy][txx] = f;
            T[2][tyy][txx] = r * r;
            T[3][tyy][txx] = f * f;
            T[4][tyy][txx] = r * f;
        }
        __syncthreads();

        const int slot = step % 5;

        // ---- per quantity: 2D box sum via WMMA (x-pass then y-pass) ----
        #pragma unroll
        for (int q = 0; q < NQ; ++q) {
            // ---- x-pass: U(20x16) = T(20x20) * Bx(20x16), Bx[k][i] = (i<=k<=i+4)
            v8f d0 = {0.f,0.f,0.f,0.f,0.f,0.f,0.f,0.f};   // rows 0..15
            v8f d1 = {0.f,0.f,0.f,0.f,0.f,0.f,0.f,0.f};   // rows 16..19 (+zero pad)
            #pragma unroll
            for (int ks = 0; ks < 5; ++ks) {
                const int kk = 4 * ks + 2 * lh;           // first K of this lane's pair
                // band B fragment: K row = kk + v, col = l16
                v2f bfrag;
                #pragma unroll
                for (int v = 0; v < 2; ++v) {
                    const int k = kk + v;
                    bfrag[v] = (k >= l16 && k <= l16 + 4) ? 1.0f : 0.0f;
                }
                // A fragments from T (contiguous pair -> ds_load_b64)
                v2f a0 = *(const v2f*)&T[q][l16][kk];
                v2f a1 = *(const v2f*)&T[q][16 + l16][kk];   // rows >=20 are zero pad
                d0 = __builtin_amdgcn_wmma_f32_16x16x4_f32(false, a0, false, bfrag,
                                                           (short)0, d0, false, false);
                d1 = __builtin_amdgcn_wmma_f32_16x16x4_f32(false, a1, false, bfrag,
                                                           (short)0, d1, false, false);
            }
            // store U transposed: Ut[col=l16][row], lane's 8 rows contiguous -> b128
            {
                v4f lo = {d0[0], d0[1], d0[2], d0[3]};
                v4f hi = {d0[4], d0[5], d0[6], d0[7]};
                *(v4f*)&Ut[q][l16][8 * lh + 0] = lo;
                *(v4f*)&Ut[q][l16][8 * lh + 4] = hi;
                v4f lo1 = {d1[0], d1[1], d1[2], d1[3]};
                v4f hi1 = {d1[4], d1[5], d1[6], d1[7]};
                *(v4f*)&Ut[q][l16][16 + 8 * lh + 0] = lo1;   // rows 24..31 = pad
                *(v4f*)&Ut[q][l16][16 + 8 * lh + 4] = hi1;
            }

            // ---- y-pass: S(16x16) = By(16x20) * U(20x16), By[j][r] = (j<=r<=j+4)
            v8f s = {0.f,0.f,0.f,0.f,0.f,0.f,0.f,0.f};
            #pragma unroll
            for (int ks = 0; ks < 5; ++ks) {
                const int rr = 4 * ks + 2 * lh;
                v2f afrag;
                #pragma unroll
                for (int v = 0; v < 2; ++v) {
                    const int r = rr + v;
                    afrag[v] = (r >= l16 && r <= l16 + 4) ? 1.0f : 0.0f;  // A row M=l16
                }
                v2f bfrag = *(const v2f*)&Ut[q][l16][rr];    // contiguous -> b64
                s = __builtin_amdgcn_wmma_f32_16x16x4_f32(false, afrag, false, bfrag,
                                                          (short)0, s, false, false);
            }

            // ---- z running window via v4f ring: Acc += S_new - Ring[slot]
            {
                v4f oldl = Ring[q][slot][lane * 2 + 0];
                v4f oldh = Ring[q][slot][lane * 2 + 1];
                #pragma unroll
                for (int g = 0; g < 4; ++g) acc[q][g]     += s[g]     - oldl[g];
                #pragma unroll
                for (int g = 0; g < 4; ++g) acc[q][4 + g] += s[4 + g] - oldh[g];
                v4f nl = {s[0], s[1], s[2], s[3]};
                v4f nh = {s[4], s[5], s[6], s[7]};
                Ring[q][slot][lane * 2 + 0] = nl;
                Ring[q][slot][lane * 2 + 1] = nh;
            }
        }

        // ---- emit cc for output slice zo = zi - 2 once window is full ----
        if (step >= 4) {
            #pragma unroll
            for (int g = 0; g < 8; ++g) {
                const float Is  = acc[0][g];
                const float Js  = acc[1][g];
                const float I2s = acc[2][g];
                const float J2s = acc[3][g];
                const float IJs = acc[4][g];
                const float uI = Is * INV_WIN;
                const float uJ = Js * INV_WIN;
                const float cross = IJs - uJ * Is - uI * Js + uI * uJ * WIN_SZ;
                const float Ivar  = I2s - 2.0f * uI * Is + uI * uI * WIN_SZ;
                const float Jvar  = J2s - 2.0f * uJ * Js + uJ * uJ * WIN_SZ;
                ccsum += (cross * cross) / (Ivar * Jvar + 1e-5f);
            }
        }
    }

    // wave32 reduction
    #pragma unroll
    for (int off = 16; off > 0; off >>= 1)
        ccsum += __shfl_down(ccsum, off, 32);

    if (lane == 0) {
        const int bid = (blockIdx.z * gridDim.y + blockIdx.y) * gridDim.x + blockIdx.x;
        partial[bid] = ccsum;
    }
}

__global__ __launch_bounds__(256) void ncc_reduce_kernel(
    const float* __restrict__ partial, int n, float* __restrict__ out)
{
    __shared__ float sm[256];
    float s = 0.0f;
    for (int i = threadIdx.x; i < n; i += 256) s += partial[i];
    sm[threadIdx.x] = s;
    __syncthreads();
    for (int w = 128; w > 0; w >>= 1) {
        if ((int)threadIdx.x < w) sm[threadIdx.x] += sm[threadIdx.x + w];
        __syncthreads();
    }
    if (threadIdx.x == 0) out[0] = sm[0] / N_VOXEL;
}

extern "C" void kernel_launch(void* const* d_in, const int* in_sizes, int n_in,
                              void* d_out, int out_size, void* d_ws, size_t ws_size,
                              hipStream_t stream)
{
    const float* real = (const float*)d_in[0];
    const float* fake = (const float*)d_in[1];
    float* out = (float*)d_out;
    float* partial = (float*)d_ws;   // 1000 floats

    dim3 grid(DHW / TILE, DHW / TILE, 2 * (DHW / ZCHUNK));   // 10 x 10 x 10
    dim3 block(32);
    ncc_tile_kernel<<<grid, block, 0, stream>>>(real, fake, partial);

    const int npart = (DHW / TILE) * (DHW / TILE) * 2 * (DHW / ZCHUNK);
    ncc_reduce_kernel<<<1, 256, 0, stream>>>(partial, npart, out);
}